// Self_WriteMPNNlayer_29661044146423
// MI455X (gfx1250) — compile-verified
//
#include <hip/hip_runtime.h>

#define N_NODES 100000
#define N_EDGES 3200000
#define D 64      // D_NM == D_EM == D_F
#define DC 128    // concat dim (D_NM + D_EM)

typedef __attribute__((ext_vector_type(2))) float v2f;
typedef __attribute__((ext_vector_type(8))) float v8f;

// ---------------- Pass 0: zero the [N,128] accumulator ----------------
__global__ void zero_acc_kernel(float4* __restrict__ acc, int n4) {
    int i = blockIdx.x * blockDim.x + threadIdx.x;
    if (i < n4) acc[i] = float4{0.f, 0.f, 0.f, 0.f};
}

// ---------------- Pass 1: scatter-add concat(nm[src], em[e]) into acc[dst] ----
// 64 threads per edge, float2 per thread: threads 0..31 -> nm row (L2-resident
// gather, coalesced within row), threads 32..63 -> em row (pure stream, b64
// loads).  Feature index = 2f for both halves; atomics land on contiguous
// f32 slots -> coalesced L2 atomic bursts via global_atomic_add_f32.
__global__ void __launch_bounds__(256)
scatter_edges_kernel(const float2* __restrict__ nm2,
                     const float2* __restrict__ em2,
                     const int*    __restrict__ esrc,
                     const int*    __restrict__ edst,
                     float*        __restrict__ acc) {
    long long idx = (long long)blockIdx.x * blockDim.x + threadIdx.x;
    int e = (int)(idx >> 6);
    int f = (int)(idx & 63);
    if (e >= N_EDGES) return;
    int src = esrc[e];
    int dst = edst[e];
    float2 v = (f < 32) ? nm2[(long long)src * 32 + f]
                        : em2[(long long)e * 32 + (f - 32)];
    float* p = acc + (long long)dst * DC + 2 * f;
    unsafeAtomicAdd(p,     v.x);
    unsafeAtomicAdd(p + 1, v.y);
}

// ---------------- Pass 2: [N,128] @ [128,64] + b + af, LeakyReLU(0.1) -------
// One wave per 32-node tile (two 16-row WMMA tiles sharing B fragments),
// 4 col-tiles of 16 -> 8 x v8f accumulators.  W staged in LDS k-pair
// interleaved so every B fragment is a single contiguous ds_load_b64.
__global__ void __launch_bounds__(256)
node_update_kernel(const float* __restrict__ acc,
                   const float* __restrict__ W,     // [128,64] row-major
                   const float* __restrict__ bias,  // [64]
                   const float* __restrict__ af,    // [N,64]
                   float*       __restrict__ out) { // [N,64]
    // Interleaved layout: element (k,col) at ((k>>1)*64 + col)*2 + (k&1)
    __shared__ float ldsW[DC * D]; // 32 KB
    for (int i = threadIdx.x; i < DC * D; i += blockDim.x) {
        int k = i >> 6, col = i & 63;
        ldsW[(((k >> 1) * D + col) << 1) | (k & 1)] = W[i];
    }
    __syncthreads();

    const int wave = threadIdx.x >> 5;
    const int lane = threadIdx.x & 31;
    const int tile = blockIdx.x * (blockDim.x >> 5) + wave;
    const int ntiles = N_NODES / 32; // 3125 (exact)
    if (tile >= ntiles) return;      // wave-uniform: EXEC all-1s inside

    const int rc   = lane & 15;          // A: row, B: column
    const int koff = (lane >> 4) << 1;   // lanes 16-31 hold k+2, k+3
    const long long node0 = (long long)tile * 32;

    v8f cA[4] = {{}, {}, {}, {}};   // rows node0    .. node0+15
    v8f cB[4] = {{}, {}, {}, {}};   // rows node0+16 .. node0+31
    const float* arow0 = acc + (node0 + rc) * DC + koff;
    const float* arow1 = arow0 + 16 * DC;
    // B fragment base: float2 at ldsW[(k+koff)*64 + col*2]
    const v2f* ldsB = (const v2f*)&ldsW[koff * D + rc * 2];

    for (int k = 0; k < DC; k += 4) {
        v2f a0 = *(const v2f*)&arow0[k];
        v2f a1 = *(const v2f*)&arow1[k];
        const v2f* bp = ldsB + (k >> 1) * D; // advance k*64 floats = k/2*64 v2f
        v2f b0 = bp[0];
        v2f b1 = bp[16];
        v2f b2 = bp[32];
        v2f b3 = bp[48];
        cA[0] = __builtin_amdgcn_wmma_f32_16x16x4_f32(false, a0, false, b0, (short)0, cA[0], false, false);
        cB[0] = __builtin_amdgcn_wmma_f32_16x16x4_f32(false, a1, false, b0, (short)0, cB[0], false, false);
        cA[1] = __builtin_amdgcn_wmma_f32_16x16x4_f32(false, a0, false, b1, (short)0, cA[1], false, false);
        cB[1] = __builtin_amdgcn_wmma_f32_16x16x4_f32(false, a1, false, b1, (short)0, cB[1], false, false);
        cA[2] = __builtin_amdgcn_wmma_f32_16x16x4_f32(false, a0, false, b2, (short)0, cA[2], false, false);
        cB[2] = __builtin_amdgcn_wmma_f32_16x16x4_f32(false, a1, false, b2, (short)0, cB[2], false, false);
        cA[3] = __builtin_amdgcn_wmma_f32_16x16x4_f32(false, a0, false, b3, (short)0, cA[3], false, false);
        cB[3] = __builtin_amdgcn_wmma_f32_16x16x4_f32(false, a1, false, b3, (short)0, cB[3], false, false);
    }

    // Epilogue: D layout -> VGPR j: rows j (lanes 0-15) / 8+j (lanes 16-31),
    // col = lane&15 within each 16-col tile.
    const int m8 = (lane >> 4) * 8;
    const float bc0 = bias[rc];
    const float bc1 = bias[16 + rc];
    const float bc2 = bias[32 + rc];
    const float bc3 = bias[48 + rc];
#pragma unroll
    for (int half = 0; half < 2; ++half) {
        const v8f* c = half ? cB : cA;
#pragma unroll
        for (int j = 0; j < 8; ++j) {
            long long row = node0 + half * 16 + m8 + j;
            const float* afr  = af  + row * D;
            float*       outr = out + row * D;
            float h;
            h = c[0][j] + bc0 + afr[rc];       outr[rc]      = (h >= 0.f) ? h : 0.1f * h;
            h = c[1][j] + bc1 + afr[16 + rc];  outr[16 + rc] = (h >= 0.f) ? h : 0.1f * h;
            h = c[2][j] + bc2 + afr[32 + rc];  outr[32 + rc] = (h >= 0.f) ? h : 0.1f * h;
            h = c[3][j] + bc3 + afr[48 + rc];  outr[48 + rc] = (h >= 0.f) ? h : 0.1f * h;
        }
    }
}

extern "C" void kernel_launch(void* const* d_in, const int* in_sizes, int n_in,
                              void* d_out, int out_size, void* d_ws, size_t ws_size,
                              hipStream_t stream) {
    const float* nm   = (const float*)d_in[0];
    const float* em   = (const float*)d_in[1];
    const float* af   = (const float*)d_in[2];
    const float* W    = (const float*)d_in[3];
    const float* b    = (const float*)d_in[4];
    const int*   esrc = (const int*)d_in[5];
    const int*   edst = (const int*)d_in[6];
    float* out = (float*)d_out;
    float* acc = (float*)d_ws;   // N_NODES * 128 floats = 51.2 MB scratch

    // Pass 0: zero accumulator (must run every call; ws is not re-poisoned)
    int n4 = N_NODES * DC / 4;
    zero_acc_kernel<<<(n4 + 255) / 256, 256, 0, stream>>>((float4*)acc, n4);

    // Pass 1: edge scatter (64 threads per edge, float2 each)
    long long total = (long long)N_EDGES * 64;
    int nblk = (int)((total + 255) / 256);
    scatter_edges_kernel<<<nblk, 256, 0, stream>>>((const float2*)nm, (const float2*)em,
                                                   esrc, edst, acc);

    // Pass 2: WMMA GEMM + bias + af + LeakyReLU
    int ntiles = N_NODES / 32;                 // 3125
    int nb2 = (ntiles + 7) / 8;                // 8 waves per block
    node_update_kernel<<<nb2, 256, 0, stream>>>(acc, W, b, af, out);
}